// GraphFlow_43722767073925
// MI455X (gfx1250) — compile-verified
//
#include <hip/hip_runtime.h>
#include <hip/hip_bf16.h>

#define THREADS 128
#define DEPTH 8

typedef int v4i __attribute__((ext_vector_type(4)));

// CDNA5 async global->LDS (ASYNCcnt) path; confirmed present on this toolchain.
#if defined(__HIP_DEVICE_COMPILE__) && \
    __has_builtin(__builtin_amdgcn_global_load_async_to_lds_b128) && \
    __has_builtin(__builtin_amdgcn_s_wait_asynccnt)
#define USE_ASYNC_LDS 1
#else
#define USE_ASYNC_LDS 0
#endif

__global__ __launch_bounds__(THREADS) void GraphFlow_43722767073925_kernel(
    const float* __restrict__ zin,   // [N,3]
    const float* __restrict__ w,     // [N,8,3,3]
    const float* __restrict__ b,     // [N,8,3]
    float* __restrict__ zout,        // [N,3]
    float* __restrict__ logdet,      // [N]
    int n)
{
  // Transposed LDS layout: [float4-index][sample] -> conflict-free ds_load_b128
  __shared__ float4 sW[18][THREADS];   // 36 KB: 72 floats of w per sample
  __shared__ float4 sB[6][THREADS];    // 12 KB: 24 floats of b per sample

  const int t = (int)threadIdx.x;
  const int chunk0 = (int)blockIdx.x * THREADS;
  int nsamp = n - chunk0;
  if (nsamp > THREADS) nsamp = THREADS;

  const float4* gw = (const float4*)w + (size_t)chunk0 * 18u;
  const float4* gb = (const float4*)b + (size_t)chunk0 * 6u;
  const int nw4 = nsamp * 18;
  const int nb4 = nsamp * 6;

#if USE_ASYNC_LDS
  for (int idx = t; idx < nw4; idx += THREADS) {      // coalesced: lane-contiguous float4s
    const int s = idx / 18;
    const int j = idx - s * 18;
    __builtin_amdgcn_global_load_async_to_lds_b128(
        (__attribute__((address_space(1))) v4i*)(gw + idx),
        (__attribute__((address_space(3))) v4i*)&sW[j][s], 0, 0);
  }
  for (int idx = t; idx < nb4; idx += THREADS) {
    const int s = idx / 6;
    const int k = idx - s * 6;
    __builtin_amdgcn_global_load_async_to_lds_b128(
        (__attribute__((address_space(1))) v4i*)(gb + idx),
        (__attribute__((address_space(3))) v4i*)&sB[k][s], 0, 0);
  }
  __builtin_amdgcn_s_wait_asynccnt(0);
  __syncthreads();
#else
  for (int idx = t; idx < nw4; idx += THREADS) {
    const int s = idx / 18;
    const int j = idx - s * 18;
    sW[j][s] = gw[idx];
  }
  for (int idx = t; idx < nb4; idx += THREADS) {
    const int s = idx / 6;
    const int k = idx - s * 6;
    sB[k][s] = gb[idx];
  }
  __syncthreads();
#endif

  if (t >= nsamp) return;

  const size_t gi = (size_t)chunk0 + (size_t)t;
  // Streaming data: non-temporal loads/stores (TH=NT) so the 28 MB of z/out
  // traffic doesn't displace the streamed w/b lines in WGP$/L2.
  float z0 = __builtin_nontemporal_load(zin + gi * 3 + 0);
  float z1 = __builtin_nontemporal_load(zin + gi * 3 + 1);
  float z2 = __builtin_nontemporal_load(zin + gi * 3 + 2);
  float ld = 0.0f;

#pragma unroll
  for (int d = 0; d < DEPTH; ++d) {
    // 9 w-floats for this depth live in 3 aligned float4s starting at (9d)>>2
    const int f0 = 9 * d, j0 = f0 >> 2, r = f0 & 3;
    float4 q0 = sW[j0 + 0][t];
    float4 q1 = sW[j0 + 1][t];
    float4 q2 = sW[j0 + 2][t];
    float win[12] = {q0.x, q0.y, q0.z, q0.w,
                     q1.x, q1.y, q1.z, q1.w,
                     q2.x, q2.y, q2.z, q2.w};
    const float w00 = win[r + 0], w01 = win[r + 1], w02 = win[r + 2];
    const float w10 = win[r + 3], w11 = win[r + 4], w12 = win[r + 5];
    const float w20 = win[r + 6], w21 = win[r + 7], w22 = win[r + 8];

    const int g0 = 3 * d, k0 = g0 >> 2, rb = g0 & 3;
    float4 p0 = sB[k0][t];
    float4 p1 = (rb >= 2) ? sB[k0 + 1][t] : p0;   // compile-time constant select
    float bwin[8] = {p0.x, p0.y, p0.z, p0.w, p1.x, p1.y, p1.z, p1.w};
    const float b0 = bwin[rb + 0], b1 = bwin[rb + 1], b2 = bwin[rb + 2];

    // d_log_det contribution: log|0.1+w00| + log|0.1+w11| + log|0.1+w22|
    ld += __logf(fabsf((0.1f + w00) * (0.1f + w11) * (0.1f + w22)));

    const float d0 = fabsf(w00) + 0.1f;
    const float d1 = fabsf(w11) + 0.1f;
    const float d2 = fabsf(w22) + 0.1f;

    // W = L @ (D*U), L unit-lower, U unit-upper, row i of U scaled by d_i
    const float W01 = d0 * w01;
    const float W02 = d0 * w02;
    const float d1w12 = d1 * w12;
    const float W10 = w10 * d0;
    const float W11 = fmaf(w10, W01, d1);
    const float W12 = fmaf(w10, W02, d1w12);
    const float W20 = w20 * d0;
    const float W21 = fmaf(w20, W01, w21 * d1);
    const float W22 = fmaf(w20, W02, fmaf(w21, d1w12, d2));

    // z_new = z @ W + b  (row-vector times matrix)
    const float n0 = fmaf(z0, d0,  fmaf(z1, W10, fmaf(z2, W20, b0)));
    const float n1 = fmaf(z0, W01, fmaf(z1, W11, fmaf(z2, W21, b1)));
    const float n2 = fmaf(z0, W02, fmaf(z1, W12, fmaf(z2, W22, b2)));

    // z = z_new @ perms[d % 3]; perms = [P_XY, P_YZ, P_XZ]
    const int pm = d % 3;
    if (pm == 0)      { z0 = n1; z1 = n0; z2 = n2; }  // P_XY: swap x,y
    else if (pm == 1) { z0 = n2; z1 = n0; z2 = n1; }  // P_YZ: (n2,n0,n1)
    else              { z0 = n2; z1 = n1; z2 = n0; }  // P_XZ: swap x,z
  }

  __builtin_nontemporal_store(z0, zout + gi * 3 + 0);
  __builtin_nontemporal_store(z1, zout + gi * 3 + 1);
  __builtin_nontemporal_store(z2, zout + gi * 3 + 2);
  __builtin_nontemporal_store(ld, logdet + gi);
}

extern "C" void kernel_launch(void* const* d_in, const int* in_sizes, int n_in,
                              void* d_out, int out_size, void* d_ws, size_t ws_size,
                              hipStream_t stream) {
  (void)n_in; (void)out_size; (void)d_ws; (void)ws_size;
  const float* z = (const float*)d_in[0];   // [N,3]
  const float* w = (const float*)d_in[1];   // [N,8,3,3]
  const float* b = (const float*)d_in[2];   // [N,8,3]
  const int n = in_sizes[0] / 3;

  float* out = (float*)d_out;
  float* zout = out;                         // first 3N floats
  float* logd = out + (size_t)n * 3;         // next N floats

  const int blocks = (n + THREADS - 1) / THREADS;
  hipLaunchKernelGGL(GraphFlow_43722767073925_kernel,
                     dim3(blocks), dim3(THREADS), 0, stream,
                     z, w, b, zout, logd, n);
}